// JSD_18622978195890
// MI455X (gfx1250) — compile-verified
//
#include <hip/hip_runtime.h>

// ---------------------------------------------------------------------------
// JSD (beta = 0.5) over rows of (N=2048, V=32000) f32 log-prob matrices.
// CDNA5 design:
//   * async global->LDS DMA stream (GLOBAL_LOAD_ASYNC_TO_LDS_B128, ASYNCcnt)
//     with a 4-deep ring buffer per wave  -> saturates HBM (23.3 TB/s floor
//     for the 524 MB single-pass read ~= 22.5 us)
//   * minimal-transcendental math: 3x v_exp/v_log_f32 per element
//   * final cross-wave reduction on the matrix pipe: one
//     V_WMMA_F32_16X16X4_F32 with an all-ones B sums the 16 wave partials.
// ---------------------------------------------------------------------------

#define JSD_BETA 0.5f

constexpr int V      = 32000;          // vocab (row length)
constexpr int LANES  = 32;             // wave32
constexpr int WAVES  = 16;             // 512 threads / block
constexpr int TILE   = LANES * 4;      // 128 floats per tile per tensor (b128/lane)
constexpr int NBUF   = 4;              // ring depth (3 tiles in flight ahead)
constexpr int TILES  = V / TILE;       // 250 tiles per row (exact)
constexpr int TILE_B = TILE * 4;       // 512 bytes per tile per tensor

typedef __attribute__((ext_vector_type(2))) float v2f;
typedef __attribute__((ext_vector_type(8))) float v8f;

// Issue one async DMA: copies this lane's 16 bytes from global byte offset goff past sbase to the LDS address in lds_off.
// GVS addressing: 64-bit SGPR base + per-lane 32-bit byte offset.
__device__ __forceinline__ void async_tile_b128(uint32_t lds_off, uint32_t goff,
                                                const float* sbase) {
    asm volatile("global_load_async_to_lds_b128 %0, %1, %2"
                 :: "v"(lds_off), "v"(goff), "s"(sbase)
                 : "memory");
}

// Wait until ASYNCcnt <= n (async loads complete in order). n in {0,2,4}.
__device__ __forceinline__ void wait_async_le(int n) {
    if (n >= 4)      asm volatile("s_wait_asynccnt 0x4" ::: "memory");
    else if (n >= 2) asm volatile("s_wait_asynccnt 0x2" ::: "memory");
    else             asm volatile("s_wait_asynccnt 0x0" ::: "memory");
}

// Drain this wave's DS pipeline before the DMA may overwrite a ring buffer.
__device__ __forceinline__ void wait_ds0() {
    asm volatile("s_wait_dscnt 0x0" ::: "memory");
}

// Low 32 bits of a flat pointer into LDS == byte offset in the group segment
// (ISA: LDS aperture maps addr[31:0] directly).
__device__ __forceinline__ uint32_t lds_off32(const void* p) {
    return (uint32_t)(uintptr_t)p;
}

// One element of the JSD sum: 3 transcendentals, rest fma/add.
__device__ __forceinline__ void jsd_accum(float lp, float lq,
                                          float& accP, float& accQ) {
    float p  = __expf(lp);                    // v_exp_f32 (+1 mul)
    float q  = __expf(lq);                    // v_exp_f32 (+1 mul)
    float m  = fmaf(JSD_BETA, p - q, q);      // lerp(q, p, beta)
    float lm = __logf(m);                     // v_log_f32 (+1 mul)
    accP = fmaf(p, lp - lm, accP);            // sum p*(lp - log_m)
    accQ = fmaf(q, lq - lm, accQ);            // sum q*(lq - log_m)
}

__global__ __launch_bounds__(WAVES * LANES)
void jsd_kernel(const float* __restrict__ log_q,
                const float* __restrict__ log_p,
                float* __restrict__ out) {
    __shared__ float ldsP[WAVES][NBUF][TILE];   // 32 KB
    __shared__ float ldsQ[WAVES][NBUF][TILE];   // 32 KB
    __shared__ float wred[WAVES];

    const int row  = blockIdx.x;
    const int tid  = threadIdx.x;
    const int wave = tid >> 5;
    const int lane = tid & (LANES - 1);

    const float* rowP = log_p + (size_t)row * V;
    const float* rowQ = log_q + (size_t)row * V;

    const uint32_t laneByte = (uint32_t)lane * 16u;
    const uint32_t pBase = lds_off32(&ldsP[wave][0][0]) + laneByte;
    const uint32_t qBase = lds_off32(&ldsQ[wave][0][0]) + laneByte;

    // Tiles handled by this wave: t = wave, wave+16, ... (< TILES)
    const int ntiles = (TILES - wave + WAVES - 1) / WAVES;

    // ---- prologue: fill the ring NBUF-1 deep --------------------------------
    for (int k = 0; k < NBUF - 1; ++k) {
        if (k < ntiles) {
            const uint32_t goff = (uint32_t)(wave + k * WAVES) * TILE_B + laneByte;
            async_tile_b128(pBase + (uint32_t)k * TILE_B, goff, rowP);
            async_tile_b128(qBase + (uint32_t)k * TILE_B, goff, rowQ);
        }
    }

    float accP = 0.0f, accQ = 0.0f;

    // ---- steady state: wait tile k, compute, issue tile k+NBUF-1 ------------
    for (int k = 0; k < ntiles; ++k) {
        const int buf = k & (NBUF - 1);
        // tiles issued beyond k that may stay outstanding:
        const int beyond0 = ntiles - 1 - k;
        const int beyond  = beyond0 < (NBUF - 2) ? beyond0 : (NBUF - 2);
        wait_async_le(2 * beyond);              // tile k's 2 DMAs are done

        const float4 lp4 = *reinterpret_cast<const float4*>(&ldsP[wave][buf][lane * 4]);
        const float4 lq4 = *reinterpret_cast<const float4*>(&ldsQ[wave][buf][lane * 4]);
        jsd_accum(lp4.x, lq4.x, accP, accQ);
        jsd_accum(lp4.y, lq4.y, accP, accQ);
        jsd_accum(lp4.z, lq4.z, accP, accQ);
        jsd_accum(lp4.w, lq4.w, accP, accQ);

        const int tn = k + (NBUF - 1);
        if (tn < ntiles) {
            // make sure every prior DS read of the reused buffer has drained
            // before the DMA engine may overwrite it (async LDS writes are not
            // ordered against the wave's DS pipe).
            wait_ds0();
            const int buf2 = tn & (NBUF - 1);
            const uint32_t goff = (uint32_t)(wave + tn * WAVES) * TILE_B + laneByte;
            async_tile_b128(pBase + (uint32_t)buf2 * TILE_B, goff, rowP);
            async_tile_b128(qBase + (uint32_t)buf2 * TILE_B, goff, rowQ);
        }
    }

    // ---- reduction ----------------------------------------------------------
    // Loss is linear in (accP, accQ): fold beta per lane, reduce one scalar.
    float r = JSD_BETA * accP + (1.0f - JSD_BETA) * accQ;
    for (int off = 16; off > 0; off >>= 1)
        r += __shfl_down(r, off, 32);
    if (lane == 0) wred[wave] = r;
    __syncthreads();

    // Final 16-way cross-wave sum on the matrix pipe: one
    // V_WMMA_F32_16X16X4_F32 with B = ones.  A is 16x4; rows 0..3 carry the
    // 16 wave partials (A[m][k] = wred[4m+k]), rows 4..15 are zero.
    // D[m][n] = sum_k A[m][k], so lane 0 holds the four chunk sums in
    // d[0..3] (C/D layout: VGPR r, lanes 0-15 -> M=r).
    if (wave == 0) {
        const int m  = lane & 15;            // A row owned by this lane
        const int kh = (lane >> 4) << 1;     // lanes 0-15: K=0,1; 16-31: K=2,3
        float a0 = 0.0f, a1 = 0.0f;
        if (m < 4) {
            a0 = wred[4 * m + kh];
            a1 = wred[4 * m + kh + 1];
        }
        v2f A = {a0, a1};
        v2f B = {1.0f, 1.0f};                // all-ones: B layout-agnostic
        v8f C = {};
        v8f D = __builtin_amdgcn_wmma_f32_16x16x4_f32(
            /*neg_a=*/false, A, /*neg_b=*/false, B,
            /*c_mod=*/(short)0, C, /*reuse_a=*/false, /*reuse_b=*/false);
        if (lane == 0)
            out[row] = D[0] + D[1] + D[2] + D[3];
    }
}

extern "C" void kernel_launch(void* const* d_in, const int* in_sizes, int n_in,
                              void* d_out, int out_size, void* d_ws, size_t ws_size,
                              hipStream_t stream) {
    // setup_inputs order: {"log_q", "log_p"}, both f32; output f32 per row.
    const float* log_q = (const float*)d_in[0];
    const float* log_p = (const float*)d_in[1];
    float* out = (float*)d_out;

    const int rows = in_sizes[0] / V;   // 2048
    jsd_kernel<<<dim3(rows), dim3(WAVES * LANES), 0, stream>>>(log_q, log_p, out);
}